// MappedMaxPool_58463094833215
// MI455X (gfx1250) — compile-verified
//
#include <hip/hip_runtime.h>

#define B_  4
#define C_  64
#define BC  256
#define H_  512
#define W_  512
#define HW  (H_ * W_)
#define OH  256
#define OW  256
#define KS  9
#define TILE 16
#define BC_PER_BLOCK 8

typedef unsigned int uint32;
typedef unsigned int v4u __attribute__((ext_vector_type(4)));
typedef int          v4i __attribute__((ext_vector_type(4)));
typedef int          v8i __attribute__((ext_vector_type(8)));
typedef float        f2  __attribute__((ext_vector_type(2)));

__global__ __launch_bounds__(256) void MappedMaxPool_58463094833215_kernel(
    const float* __restrict__ xin,
    const float* __restrict__ smap,
    float* __restrict__ out) {
  // Raw sample_map tile: 16 rows x (16 positions * 9 samples * 2 coords) = 4608 f32 = 18 KB
  __shared__ float s_raw[TILE * TILE * KS * 2];

  const int tid    = threadIdx.x;
  const int tileId = blockIdx.x;
  const int oh0    = (tileId / (OW / TILE)) * TILE;
  const int ow0    = (tileId % (OW / TILE)) * TILE;
  const int bc0    = blockIdx.y * BC_PER_BLOCK;

  // ---- TDM: async DMA of the sample_map tile into LDS (one wave issues) ----
  if (tid < 32) {
    unsigned long long gaddr =
        (unsigned long long)(size_t)smap +
        (unsigned long long)((oh0 * OW + ow0) * (KS * 2)) * 4ull;
    uint32 lds_addr = (uint32)(size_t)(&s_raw[0]);

    v4u g0;  v8i g1;  v4i g2;  v4i g3;  v8i g4;
    // D# group 0: count=1 (valid), lds_addr, global_addr[56:0], type=2 ("image")
    g0[0] = 1u;
    g0[1] = lds_addr;
    g0[2] = (uint32)gaddr;
    g0[3] = (uint32)((gaddr >> 32) & 0x01FFFFFFu) | (2u << 30);
    // D# group 1: data_size=4B(code 2); tensor_dim0=4608, tensor_dim1=256;
    //             tile_dim0=288, tile_dim1=16; tensor_dim0_stride=4608
    g1[0] = (2 << 16);
    g1[1] = (int)((4608u & 0xFFFFu) << 16);
    g1[2] = (int)((4608u >> 16) | ((256u & 0xFFFFu) << 16));
    g1[3] = (int)((256u >> 16) | (288u << 16));
    g1[4] = 16;
    g1[5] = 4608;
    g1[6] = 0;
    g1[7] = 0;
    g2[0] = g2[1] = g2[2] = g2[3] = 0;   // 2D tensor: groups 2/3 unused
    g3[0] = g3[1] = g3[2] = g3[3] = 0;
    g4[0] = g4[1] = g4[2] = g4[3] = g4[4] = g4[5] = g4[6] = g4[7] = 0;
    __builtin_amdgcn_tensor_load_to_lds(g0, g1, g2, g3, g4, 0);
    __builtin_amdgcn_s_wait_tensorcnt(0);
  }
  __syncthreads();

  // ---- decode this thread's 9 samples into registers ----
  int   offs[KS];
  float wxs[KS], wys[KS];
  {
    const float* pr = &s_raw[tid * (KS * 2)];
#pragma unroll
    for (int k = 0; k < KS; ++k) {
      float sx = pr[2 * k + 0];
      float sy = pr[2 * k + 1];
      int x0 = (int)floorf(sx);
      int y0 = (int)floorf(sy);
      x0 = x0 < 0 ? 0 : (x0 > W_ - 2 ? W_ - 2 : x0);
      y0 = y0 < 0 ? 0 : (y0 > H_ - 2 ? H_ - 2 : y0);
      wxs[k]  = sx - (float)x0;   // fraction vs clamped corner (x1 = x0+1 valid)
      wys[k]  = sy - (float)y0;
      offs[k] = y0 * W_ + x0;
    }
  }

  const int    orow = oh0 + (tid >> 4);
  const int    ocol = ow0 + (tid & 15);
  const size_t opix = (size_t)orow * OW + ocol;

  // ---- loop over 8 channel images; coords amortized, images L2-resident ----
  for (int i = 0; i < BC_PER_BLOCK; ++i) {
    const int bc = bc0 + i;
    const float* img = xin + (size_t)bc * HW;
    float m = -3.402823466e38f;
#pragma unroll
    for (int k = 0; k < KS; ++k) {
      f2 a, b;
      __builtin_memcpy(&a, img + offs[k], 8);        // row y0: x0, x0+1
      __builtin_memcpy(&b, img + offs[k] + W_, 8);   // row y1: x0, x0+1
      const float wx = wxs[k], wy = wys[k];
      const float top = (1.0f - wx) * a.x + wx * a.y;
      const float bot = (1.0f - wx) * b.x + wx * b.y;
      const float v   = (1.0f - wy) * top + wy * bot;
      m = fmaxf(m, v);
    }
    // Non-temporal: output is write-once; keep L2 for gather-resident images.
    __builtin_nontemporal_store(m, out + (size_t)bc * (OH * OW) + opix);
  }
}

extern "C" void kernel_launch(void* const* d_in, const int* in_sizes, int n_in,
                              void* d_out, int out_size, void* d_ws, size_t ws_size,
                              hipStream_t stream) {
  const float* x    = (const float*)d_in[0];   // [4,64,512,512] f32
  const float* smap = (const float*)d_in[1];   // [256,256,9,2]  f32
  float*       out  = (float*)d_out;           // [4,64,256,256] f32
  (void)in_sizes; (void)n_in; (void)out_size; (void)d_ws; (void)ws_size;

  dim3 grid((OH / TILE) * (OW / TILE), BC / BC_PER_BLOCK);  // (256, 32)
  MappedMaxPool_58463094833215_kernel<<<grid, 256, 0, stream>>>(x, smap, out);
}